// GPUQuantumCircuit_5884105195559
// MI455X (gfx1250) — compile-verified
//
#include <hip/hip_runtime.h>

typedef float v2f __attribute__((ext_vector_type(2)));
typedef float v8f __attribute__((ext_vector_type(8)));

__device__ __forceinline__ float2 cmul(float2 a, float2 b) {
    return make_float2(a.x * b.x - a.y * b.y, a.x * b.y + a.y * b.x);
}
__device__ __forceinline__ float2 cadd(float2 a, float2 b) {
    return make_float2(a.x + b.x, a.y + b.y);
}

#define NQ   14
#define DIM  16384
#define NT   256

// LDS layout (SoA, floats):
//   re   : 16384          (state real plane)
//   im   : 16384          (state imag plane)
//   Gr   :   256          (16x16 fused gate, real)
//   Gi   :   256          (16x16 fused gate, imag)
//   Gin  :   256          (16x16 fused gate, -imag  -> f32 WMMA has no A-neg)
//   Uq   :  14*4 float2   (fused per-qubit 2x2 complex gates)
//   enc  :  14*2          (encoding cos/sin per qubit)
//   lo8  :   256          (product table bits 0-7)
//   hi6  :    64          (product table bits 8-13)
#define SMEM_FLOATS (16384 + 16384 + 256 + 256 + 256)
#define SMEM_BYTES  (SMEM_FLOATS * 4 + 56 * 8 + (28 + 256 + 64) * 4)

__global__ __launch_bounds__(NT) void qsim14_kernel(
    const float* __restrict__ latent,   // [256][14]
    const float* __restrict__ weights,  // [84]
    float* __restrict__ out)            // [256][16384]
{
    extern __shared__ unsigned char smem_raw[];
    float*  re  = reinterpret_cast<float*>(smem_raw);
    float*  im  = re + DIM;
    float*  Gr  = im + DIM;
    float*  Gi  = Gr + 256;
    float*  Gin = Gi + 256;
    float2* Uq  = reinterpret_cast<float2*>(Gin + 256);   // 8-byte aligned (offset 134144)
    float*  enc = reinterpret_cast<float*>(Uq + NQ * 4);
    float*  lo8 = enc + NQ * 2;
    float*  hi6 = lo8 + 256;

    const int tid  = threadIdx.x;
    const int bidx = blockIdx.x;
    const int lane = tid & 31;
    const int wid  = tid >> 5;
    const int hi   = lane >> 4;   // 0: lanes 0-15, 1: lanes 16-31
    const int l16  = lane & 15;

    // ---------- encoding: product-state init ----------
    if (tid < NQ) {
        float half = latent[bidx * NQ + tid] * 1.57079632679489662f;  // (latent*pi)/2
        float s, c;
        sincosf(half, &s, &c);
        enc[tid * 2 + 0] = c;
        enc[tid * 2 + 1] = s;
    }
    __syncthreads();
    {
        float p = 1.0f;
        #pragma unroll
        for (int k = 0; k < 8; ++k) p *= enc[k * 2 + ((tid >> k) & 1)];
        lo8[tid] = p;
        if (tid < 64) {
            float q = 1.0f;
            #pragma unroll
            for (int k = 0; k < 6; ++k) q *= enc[(8 + k) * 2 + ((tid >> k) & 1)];
            hi6[tid] = q;
        }
    }
    __syncthreads();
    #pragma unroll 4
    for (int i = 0; i < 16; ++i) {
        int n = i * 1024 + tid * 4;
        float h = hi6[n >> 8];           // same high bits for all 4
        float4 r;
        r.x = lo8[(n + 0) & 255] * h;
        r.y = lo8[(n + 1) & 255] * h;
        r.z = lo8[(n + 2) & 255] * h;
        r.w = lo8[(n + 3) & 255] * h;
        *reinterpret_cast<float4*>(re + n) = r;
        *reinterpret_cast<float4*>(im + n) = make_float4(0.f, 0.f, 0.f, 0.f);
    }
    __syncthreads();

    // ---------- layers ----------
    for (int layer = 0; layer < 2; ++layer) {
        // fused U_q = RY(g) * RZ(b) * RY(a), one qubit per thread
        if (tid < NQ) {
            int base = layer * (NQ * 3) + tid * 3;
            float al = weights[base], be = weights[base + 1], ga = weights[base + 2];
            float sa, ca, sb, cb, sg, cg;
            sincosf(0.5f * al, &sa, &ca);
            sincosf(0.5f * be, &sb, &cb);
            sincosf(0.5f * ga, &sg, &cg);
            float2 e0 = make_float2(cb, -sb);   // exp(-i b/2)
            float2 e1 = make_float2(cb,  sb);   // exp(+i b/2)
            Uq[tid * 4 + 0] = make_float2( cg*ca*e0.x - sg*sa*e1.x,  cg*ca*e0.y - sg*sa*e1.y);
            Uq[tid * 4 + 1] = make_float2(-cg*sa*e0.x - sg*ca*e1.x, -cg*sa*e0.y - sg*ca*e1.y);
            Uq[tid * 4 + 2] = make_float2( sg*ca*e0.x + cg*sa*e1.x,  sg*ca*e0.y + cg*sa*e1.y);
            Uq[tid * 4 + 3] = make_float2(-sg*sa*e0.x + cg*ca*e1.x, -sg*sa*e0.y + cg*ca*e1.y);
        }
        __syncthreads();

        // groups of 4 qubits -> 16x16 complex matrix -> WMMA f32 16x16x4
        for (int g = 0; g < 3; ++g) {
            { // build Kronecker matrix: one entry per thread (SoA planes)
                int a = tid >> 4, j = tid & 15;
                float2 p = make_float2(1.0f, 0.0f);
                #pragma unroll
                for (int k = 0; k < 4; ++k) {
                    float2 u = Uq[(g * 4 + k) * 4 + ((a >> k) & 1) * 2 + ((j >> k) & 1)];
                    p = cmul(p, u);
                }
                Gr[tid]  = p.x;
                Gi[tid]  = p.y;
                Gin[tid] = -p.y;
            }
            __syncthreads();

            const int b = g * 4;                 // bit base of group
            const int lomask = (1 << b) - 1;

            // A operand regs (uniform across tiles): ISA 16x4 f32 layout,
            // adjacent K pairs -> direct 8-byte LDS loads, no repacking
            v2f Ar[4], Ai[4], Ain[4];
            #pragma unroll
            for (int k = 0; k < 4; ++k) {
                int idx = l16 * 16 + 4 * k + 2 * hi;
                Ar[k]  = *reinterpret_cast<v2f*>(Gr  + idx);
                Ai[k]  = *reinterpret_cast<v2f*>(Gi  + idx);
                Ain[k] = *reinterpret_cast<v2f*>(Gin + idx);
            }

            // 64 column-tiles of 16, 8 per wave; literal trip count -> scalar loop,
            // EXEC statically all-ones around WMMA
            for (int it = 0; it < 8; ++it) {
                const int t = it * 8 + wid;
                int c = t * 16 + l16;
                int nbase = ((c >> b) << (b + 4)) | (c & lomask);
                v8f Dr = {};
                v8f Di = {};
                #pragma unroll
                for (int k = 0; k < 4; ++k) {
                    int j0 = 4 * k + 2 * hi;
                    int n0 = nbase + (j0 << b);
                    int n1 = n0 + (1 << b);
                    v2f Br = {re[n0], re[n1]};
                    v2f Bi = {im[n0], im[n1]};
                    // Dr += Ur*Br - Ui*Bi ;  Di += Ur*Bi + Ui*Br
                    Dr = __builtin_amdgcn_wmma_f32_16x16x4_f32(false, Ar[k],  false, Br, (short)0, Dr, false, false);
                    Dr = __builtin_amdgcn_wmma_f32_16x16x4_f32(false, Ain[k], false, Bi, (short)0, Dr, false, false);
                    Di = __builtin_amdgcn_wmma_f32_16x16x4_f32(false, Ar[k],  false, Bi, (short)0, Di, false, false);
                    Di = __builtin_amdgcn_wmma_f32_16x16x4_f32(false, Ai[k],  false, Br, (short)0, Di, false, false);
                }
                #pragma unroll
                for (int v = 0; v < 8; ++v) {
                    int a = v + hi * 8;
                    int n = nbase + (a << b);
                    re[n] = Dr[v];
                    im[n] = Di[v];
                }
            }
            __syncthreads();
        }

        // qubits 12 and 13: plain complex butterflies
        for (int q = 12; q <= 13; ++q) {
            float2 u00 = Uq[q * 4 + 0], u01 = Uq[q * 4 + 1];
            float2 u10 = Uq[q * 4 + 2], u11 = Uq[q * 4 + 3];
            for (int p = tid; p < 8192; p += NT) {
                int n0 = ((p >> q) << (q + 1)) | (p & ((1 << q) - 1));
                int n1 = n0 | (1 << q);
                float2 a0 = make_float2(re[n0], im[n0]);
                float2 a1 = make_float2(re[n1], im[n1]);
                float2 r0 = cadd(cmul(u00, a0), cmul(u01, a1));
                float2 r1 = cadd(cmul(u10, a0), cmul(u11, a1));
                re[n0] = r0.x; im[n0] = r0.y;
                re[n1] = r1.x; im[n1] = r1.y;
            }
            __syncthreads();
        }

        // CNOT ladder: pair (i, i+1) -> ctrl bit 13-i, tgt bit 12-i; in-place swaps
        for (int i = 0; i < 13; ++i) {
            int tb = 12 - i;
            for (int p = tid; p < 4096; p += NT) {
                int n = ((p >> tb) << (tb + 2)) | (1 << (tb + 1)) | (p & ((1 << tb) - 1));
                int m = n | (1 << tb);
                float rn = re[n], rm = re[m];
                float in_ = im[n], im_ = im[m];
                re[n] = rm; re[m] = rn;
                im[n] = im_; im[m] = in_;
            }
            __syncthreads();
        }
    }

    // ---------- probabilities out (b128 LDS reads + b128 global stores) ----------
    float* o = out + (size_t)bidx * DIM;
    #pragma unroll 4
    for (int i = 0; i < 16; ++i) {
        int n = i * 1024 + tid * 4;
        float4 r = *reinterpret_cast<float4*>(re + n);
        float4 m = *reinterpret_cast<float4*>(im + n);
        float4 p;
        p.x = r.x * r.x + m.x * m.x;
        p.y = r.y * r.y + m.y * m.y;
        p.z = r.z * r.z + m.z * m.z;
        p.w = r.w * r.w + m.w * m.w;
        *reinterpret_cast<float4*>(o + n) = p;
    }
}

extern "C" void kernel_launch(void* const* d_in, const int* in_sizes, int n_in,
                              void* d_out, int out_size, void* d_ws, size_t ws_size,
                              hipStream_t stream) {
    const float* latent  = (const float*)d_in[0];   // 256*14 f32
    const float* weights = (const float*)d_in[1];   // 84 f32
    float* out = (float*)d_out;                     // 256*16384 f32
    (void)in_sizes; (void)n_in; (void)out_size; (void)d_ws; (void)ws_size;

    qsim14_kernel<<<dim3(256), dim3(NT), SMEM_BYTES, stream>>>(latent, weights, out);
}